// WeightAdaptiveGallinear_25400436588940
// MI455X (gfx1250) — compile-verified
//
#include <hip/hip_runtime.h>
#include <hip/hip_bf16.h>

// Problem constants (from reference)
#define IN_F   16
#define OUT_F  16
#define M_DIM  8            // NH * NEIG
#define P_DIM  272          // (IN_F+1)*OUT_F
#define LAT    1088         // P*M/2
#define BATCH  16384
#define XROW   (IN_F + LAT + 1)   // 1105 floats per x row
#define LATOFF (IN_F + 1)         // latent starts at col 17

// LDS padding: row stride 1092 (1092 % 64 == 4) -> conflict-free WMMA A reads
#define LSTRIDE 1092
#define ROWS_PER_BLOCK 16
#define THREADS 256

typedef __attribute__((ext_vector_type(2))) float v2f;
typedef __attribute__((ext_vector_type(8))) float v8f;

#if __has_builtin(__builtin_amdgcn_global_load_async_to_lds_b32)
#define USE_ASYNC_LDS 1
#else
#define USE_ASYNC_LDS 0
#endif

typedef __attribute__((address_space(1))) int as1_int;
typedef __attribute__((address_space(3))) int as3_int;

// LDS layout (floats):
//  lat   : 16*1092          latent tile
//  wlds  : 8*1092           W (8 x 1088)
//  inp   : 16*16            input tile
//  dpart : 8*256            per-wave partial WMMA D tiles
//  svals : 16*8             signed cos/sin values
#define SM_LAT   0
#define SM_W     (SM_LAT + ROWS_PER_BLOCK * LSTRIDE)
#define SM_INP   (SM_W + M_DIM * LSTRIDE)
#define SM_DP    (SM_INP + ROWS_PER_BLOCK * IN_F)
#define SM_SV    (SM_DP + 8 * 256)
#define SM_TOTAL (SM_SV + ROWS_PER_BLOCK * M_DIM)

// One 4-byte global -> LDS copy. Async path uses the CDNA5 ASYNCcnt-tracked
// DMA (global_load_async_to_lds_b32): no VGPR round-trip, no loadcnt/dscnt.
__device__ __forceinline__ void cp_to_lds_f32(float* dst, const float* src,
                                              const float* smem_base) {
#if USE_ASYNC_LDS
    unsigned off = (unsigned)((const char*)dst - (const char*)smem_base)
                 + (unsigned)__builtin_amdgcn_groupstaticsize();
    __builtin_amdgcn_global_load_async_to_lds_b32(
        (as1_int*)(unsigned long long)src,
        (as3_int*)off,
        /*imm offset*/0, /*cpol*/0);
#else
    *dst = *src;
#endif
}

__device__ __forceinline__ void wait_async_copies() {
#if USE_ASYNC_LDS
#if __has_builtin(__builtin_amdgcn_s_wait_asynccnt)
    __builtin_amdgcn_s_wait_asynccnt(0);
#else
    asm volatile("s_wait_asynccnt 0x0" ::: "memory");
#endif
#endif
}

__global__ __launch_bounds__(THREADS) void gallinear_kernel(
    const float* __restrict__ xg,   // (BATCH, 1105)
    const float* __restrict__ Wg,   // (8, 1088)
    float* __restrict__ out)        // (BATCH, 16)
{
    extern __shared__ float smem[];
    float* lat   = smem + SM_LAT;
    float* wlds  = smem + SM_W;
    float* inpl  = smem + SM_INP;
    float* dpart = smem + SM_DP;
    float* svals = smem + SM_SV;

    const int tid  = threadIdx.x;
    const int row0 = blockIdx.x * ROWS_PER_BLOCK;

    // hint: W is small (34 KB) and reused by every block -> keep it hot
    __builtin_prefetch(Wg, 0, 3);

    // ---- Phase A: global -> LDS staging (async DMA when available) -------
    // latent tile: 16 rows x 1088 cols
    for (int r = 0; r < ROWS_PER_BLOCK; ++r) {
        const float* src = xg + (size_t)(row0 + r) * XROW + LATOFF;
        for (int c = tid; c < LAT; c += THREADS)
            cp_to_lds_f32(&lat[r * LSTRIDE + c], &src[c], smem);
    }
    // W: 8 rows x 1088 cols
    for (int r = 0; r < M_DIM; ++r) {
        const float* src = Wg + (size_t)r * LAT;
        for (int c = tid; c < LAT; c += THREADS)
            cp_to_lds_f32(&wlds[r * LSTRIDE + c], &src[c], smem);
    }
    // inp tile: 16 rows x 16 cols, exactly 256 elements
    {
        int r = tid >> 4, c = tid & 15;
        cp_to_lds_f32(&inpl[tid], &xg[(size_t)(row0 + r) * XROW + c], smem);
    }
    wait_async_copies();
    __syncthreads();

    // ---- Phase B: dilation = latent @ W^T via V_WMMA_F32_16X16X4_F32 -----
    // Wave w covers K in [w*136, w*136+136), 34 WMMA steps, partial D to LDS.
    {
        const int lane = tid & 31;
        const int wave = tid >> 5;
        const int mrow = lane & 15;     // A row / B col (N)
        const int half = lane >> 4;     // K-half select within the 4-slice
        const int k0   = wave * (LAT / 8);

        v8f acc = {};
        const float* arow = lat + mrow * LSTRIDE;
        // Unconditional load from a clamped row; zero-select on the VALUE so
        // EXEC never changes inside the WMMA loop (no saveexec serialization).
        const float* wrow = wlds + (mrow & 7) * LSTRIDE;
        const bool bvalid = (mrow < M_DIM);

#pragma unroll 2
        for (int kk = 0; kk < LAT / 8; kk += 4) {
            const int k = k0 + kk + 2 * half;
            v2f a = *(const v2f*)(arow + k);      // ds_load_b64
            v2f bv = *(const v2f*)(wrow + k);     // ds_load_b64
            v2f b;
            b.x = bvalid ? bv.x : 0.0f;           // v_cndmask
            b.y = bvalid ? bv.y : 0.0f;
            // D = A(16x4 f32) * B(4x16 f32) + C
            acc = __builtin_amdgcn_wmma_f32_16x16x4_f32(
                false, a, false, b, (short)0, acc, false, false);
        }
#pragma unroll
        for (int g = 0; g < 8; ++g)
            dpart[wave * 256 + (g + 8 * half) * 16 + mrow] = acc[g];
    }
    __syncthreads();

    // ---- Phase C: reduce partials, vals = sign * cos/sin(s0 * dil) -------
    if (tid < ROWS_PER_BLOCK * M_DIM) {   // 128 threads: (row, m)
        const int r = tid >> 3;
        const int m = tid & 7;
        float dil = 0.0f;
#pragma unroll
        for (int q = 0; q < 8; ++q)
            dil += dpart[q * 256 + r * 16 + m];
        const float s0 = xg[(size_t)(BATCH - 1) * XROW + IN_F];
        const float th = s0 * dil;
        float v = (m < 4) ? __cosf(th) : __sinf(th);
        if (m == 2) v = -v;               // col_sign[2] = -1 folded in
        svals[r * M_DIM + m] = v;
    }
    __syncthreads();

    // ---- Phase D: fused w-construction + output GEMV ---------------------
    // out[b, 2j] = 0; out[b, 2j+1] = sum_i inp[i]*w[16i+o] + w[256+o]
    if (tid < ROWS_PER_BLOCK * 8) {       // 128 threads: (row, odd-o index)
        const int r = tid >> 3;
        const int j = tid & 7;
        const int o = 2 * j + 1;

        float sv[8];
#pragma unroll
        for (int m = 0; m < 8; ++m) sv[m] = svals[r * M_DIM + m];

        const float* lr = lat + r * LSTRIDE;
        float accum = 0.0f;
#pragma unroll
        for (int i = 0; i < IN_F; ++i) {
            int base = 128 * i + 8 * o;   // 8*(16i+o)
            if (base >= LAT) base -= LAT;
            float wp = 0.0f;
#pragma unroll
            for (int m = 0; m < 8; ++m)
                wp = fmaf(lr[base + m], sv[m], wp);
            accum = fmaf(inpl[r * 16 + i], wp, accum);
        }
        // bias term: p = 256 + o -> base = (2048 + 8o) - 1088 = 960 + 8o
        {
            const int base = 960 + 8 * o;
            float wp = 0.0f;
#pragma unroll
            for (int m = 0; m < 8; ++m)
                wp = fmaf(lr[base + m], sv[m], wp);
            accum += wp;
        }
        v2f o2; o2.x = 0.0f; o2.y = accum;
        *(v2f*)(out + (size_t)(row0 + r) * OUT_F + 2 * j) = o2;
    }
}

extern "C" void kernel_launch(void* const* d_in, const int* in_sizes, int n_in,
                              void* d_out, int out_size, void* d_ws, size_t ws_size,
                              hipStream_t stream) {
    const float* xg = (const float*)d_in[0];   // (16384, 1105) f32
    const float* Wg = (const float*)d_in[1];   // (8, 1088) f32
    float* out = (float*)d_out;                // (16384, 16) f32

    const size_t shmem = (size_t)SM_TOTAL * sizeof(float);  // ~112 KB
    // allow >64KB dynamic LDS (gfx1250 WGP has 320 KB)
    (void)hipFuncSetAttribute((const void*)gallinear_kernel,
                              hipFuncAttributeMaxDynamicSharedMemorySize,
                              (int)shmem);

    dim3 grid(BATCH / ROWS_PER_BLOCK);   // 1024 blocks
    dim3 block(THREADS);                 // 8 wave32 waves
    gallinear_kernel<<<grid, block, shmem, stream>>>(xg, Wg, out);
}